// EquivariantDecoder_18580028522964
// MI455X (gfx1250) — compile-verified
//
#include <hip/hip_runtime.h>
#include <math.h>

#define H 128
#define HS 136   // LDS row stride (padded: 136 % 64 banks = 8 -> half-waves hit disjoint banks)

typedef __attribute__((ext_vector_type(2))) float v2f;
typedef __attribute__((ext_vector_type(8))) float v8f;

__device__ __forceinline__ float silu_f(float t) {
    return t / (1.0f + __expf(-t));
}

// ---------------------------------------------------------------------------
// Zero the scatter workspace (gsum[N*3] ++ gcnt[N]); harness poisons ws.
// ---------------------------------------------------------------------------
__global__ void eqdec_zero_ws(float* __restrict__ p, int n) {
    int i = blockIdx.x * blockDim.x + threadIdx.x;
    if (i < n) p[i] = 0.0f;
}

// ---------------------------------------------------------------------------
// Edge kernel: w = silu(m_ij @ W1 + b1) @ W2 + b2  per edge, then scatter
// rel(x[src]-x[dst]) * w into gsum[dst], and 1.0 into gcnt[dst].
// One wave = one 16-edge tile. 4 waves / block. W1 staged in LDS.
// ---------------------------------------------------------------------------
__global__ void eqdec_edge_mlp_scatter(const float* __restrict__ m_ij,
                                       const float* __restrict__ x,
                                       const int*   __restrict__ edge_index,
                                       const float* __restrict__ W1,
                                       const float* __restrict__ b1,
                                       const float* __restrict__ W2,   // [H,1]
                                       const float* __restrict__ b2,   // [1]
                                       float* __restrict__ gsum,
                                       float* __restrict__ gcnt,
                                       int E_) {
    extern __shared__ float sW1[];  // H * HS floats
    const int tid = threadIdx.x;

    for (int i = tid; i < H * H; i += blockDim.x)
        sW1[(i >> 7) * HS + (i & 127)] = W1[i];
    __syncthreads();

    const int wave  = tid >> 5;
    const int lane  = tid & 31;
    const int col16 = lane & 15;
    const int half  = lane >> 4;
    const long e0   = ((long)blockIdx.x * 4 + wave) * 16;
    if (e0 >= E_) return;  // wave-uniform (no barriers below)

    // A fragments: row M = e0+col16, this lane covers K = {4k+2*half, 4k+2*half+1}
    const v2f* ap = (const v2f*)(m_ij + (e0 + col16) * H) + half;
    v2f a[32];
#pragma unroll
    for (int k = 0; k < 32; ++k) a[k] = ap[2 * k];

    float r[8];
#pragma unroll
    for (int i = 0; i < 8; ++i) r[i] = 0.0f;

    for (int n = 0; n < 8; ++n) {
        const int col = n * 16 + col16;        // output column N = col (lane-mapped)
        v8f c = {0.f, 0.f, 0.f, 0.f, 0.f, 0.f, 0.f, 0.f};
        const float* bp = sW1 + (2 * half) * HS + col;
#pragma unroll
        for (int k = 0; k < 32; ++k) {
            v2f b;
            b.x = bp[(4 * k) * HS];            // W1[K=4k+2*half  ][col]
            b.y = bp[(4 * k + 1) * HS];        // W1[K=4k+2*half+1][col]
            c = __builtin_amdgcn_wmma_f32_16x16x4_f32(
                    false, a[k], false, b, (short)0, c, false, false);
        }
        const float bias = b1[col];
        const float w2v  = W2[col];
#pragma unroll
        for (int i = 0; i < 8; ++i) {          // c[i]: row i (lanes 0-15) / row 8+i (lanes 16-31)
            float t = c[i] + bias;
            r[i] += silu_f(t) * w2v;
        }
    }

    // Reduce each r[i] over its 16-lane group -> w for edges (i) and (8+i).
    float wv = 0.0f;
#pragma unroll
    for (int i = 0; i < 8; ++i) {
        float s = r[i];
        s += __shfl_xor(s, 1, 32);
        s += __shfl_xor(s, 2, 32);
        s += __shfl_xor(s, 4, 32);
        s += __shfl_xor(s, 8, 32);             // groups of 16 fully reduced
        float sh = __shfl_xor(s, 16, 32);      // lanes 0-15 now see edge 8+i's sum
        if (lane == i)     wv = s;             // edge e0+i
        if (lane == i + 8) wv = sh;            // edge e0+8+i
    }

    if (lane < 16) {
        const long e  = e0 + lane;
        const float w = wv + b2[0];
        const int src = edge_index[e];
        const int dst = edge_index[(long)E_ + e];
        const float rx = x[src * 3 + 0] - x[dst * 3 + 0];
        const float ry = x[src * 3 + 1] - x[dst * 3 + 1];
        const float rz = x[src * 3 + 2] - x[dst * 3 + 2];
        atomicAdd(&gsum[dst * 3 + 0], rx * w);
        atomicAdd(&gsum[dst * 3 + 1], ry * w);
        atomicAdd(&gsum[dst * 3 + 2], rz * w);
        atomicAdd(&gcnt[dst], 1.0f);
    }
}

// ---------------------------------------------------------------------------
// Node kernel: alpha = silu(h @ vg_W1 + b1) @ vg_W2 + b2  ([N,5]),
// out = sum_j alpha_j * vel_all[:,j,:] + gsum / max(gcnt,1).
// One wave = one 16-node tile, 2 waves / block. Hidden staged in LDS.
// ---------------------------------------------------------------------------
__global__ void eqdec_node_mlp_combine(const float* __restrict__ h,
                                       const float* __restrict__ vel_all,  // [N,5,3]
                                       const float* __restrict__ W1,
                                       const float* __restrict__ b1,
                                       const float* __restrict__ W2,   // [H,5]
                                       const float* __restrict__ b2,   // [5]
                                       const float* __restrict__ gsum,
                                       const float* __restrict__ gcnt,
                                       float* __restrict__ out,
                                       int N_) {
    extern __shared__ float smem[];            // sW1[H*HS] ++ hidden[2][16*HS]
    float* sW1  = smem;
    float* sHid = smem + H * HS;
    const int tid = threadIdx.x;

    for (int i = tid; i < H * H; i += blockDim.x)
        sW1[(i >> 7) * HS + (i & 127)] = W1[i];
    __syncthreads();

    const int wave  = tid >> 5;
    const int lane  = tid & 31;
    const int col16 = lane & 15;
    const int half  = lane >> 4;
    const long n0   = ((long)blockIdx.x * 2 + wave) * 16;
    const bool active = (n0 < N_);
    float* hid = sHid + wave * 16 * HS;

    if (active) {
        const v2f* ap = (const v2f*)(h + (n0 + col16) * H) + half;
        v2f a[32];
#pragma unroll
        for (int k = 0; k < 32; ++k) a[k] = ap[2 * k];

        for (int n = 0; n < 8; ++n) {
            const int col = n * 16 + col16;
            v8f c = {0.f, 0.f, 0.f, 0.f, 0.f, 0.f, 0.f, 0.f};
            const float* bp = sW1 + (2 * half) * HS + col;
#pragma unroll
            for (int k = 0; k < 32; ++k) {
                v2f b;
                b.x = bp[(4 * k) * HS];
                b.y = bp[(4 * k + 1) * HS];
                c = __builtin_amdgcn_wmma_f32_16x16x4_f32(
                        false, a[k], false, b, (short)0, c, false, false);
            }
            const float bias = b1[col];
#pragma unroll
            for (int i = 0; i < 8; ++i) {
                float t = c[i] + bias;
                hid[(i + 8 * half) * HS + col] = silu_f(t);
            }
        }
    }
    __syncthreads();  // all waves reach this (uniform barrier count)

    if (active && lane < 16) {
        const long node = n0 + lane;
        const float* hr = hid + lane * HS;
        float alpha[5];
#pragma unroll
        for (int j = 0; j < 5; ++j) alpha[j] = b2[j];
        for (int k = 0; k < H; ++k) {
            const float hv = hr[k];
#pragma unroll
            for (int j = 0; j < 5; ++j) alpha[j] += hv * W2[k * 5 + j];
        }
        const float cn = fmaxf(gcnt[node], 1.0f);
#pragma unroll
        for (int cdim = 0; cdim < 3; ++cdim) {
            float v = 0.0f;
#pragma unroll
            for (int j = 0; j < 5; ++j)
                v += alpha[j] * vel_all[node * 15 + j * 3 + cdim];
            out[node * 3 + cdim] = v + gsum[node * 3 + cdim] / cn;
        }
    }
}

// ---------------------------------------------------------------------------
extern "C" void kernel_launch(void* const* d_in, const int* in_sizes, int n_in,
                              void* d_out, int out_size, void* d_ws, size_t ws_size,
                              hipStream_t stream) {
    const float* h_        = (const float*)d_in[0];
    const float* m_ij      = (const float*)d_in[1];
    const float* x_        = (const float*)d_in[2];
    const float* vel_all   = (const float*)d_in[3];
    const int*   edge_idx  = (const int*)  d_in[4];
    const float* ew_W1     = (const float*)d_in[5];
    const float* ew_b1     = (const float*)d_in[6];
    const float* ew_W2     = (const float*)d_in[7];
    const float* ew_b2     = (const float*)d_in[8];
    const float* vg_W1     = (const float*)d_in[9];
    const float* vg_b1     = (const float*)d_in[10];
    const float* vg_W2     = (const float*)d_in[11];
    const float* vg_b2     = (const float*)d_in[12];

    const int N = in_sizes[0] / H;     // 100000
    const int E = in_sizes[1] / H;     // 1600000

    float* gsum = (float*)d_ws;        // [N*3]
    float* gcnt = gsum + (size_t)N * 3;  // [N]

    // 1) zero scatter buffers
    {
        const int tot = N * 4;
        eqdec_zero_ws<<<(tot + 255) / 256, 256, 0, stream>>>(gsum, tot);
    }
    // 2) edge MLP + scatter (4 waves/block, 16 edges/wave)
    {
        const int blocks = (E + 63) / 64;
        const size_t shmem = (size_t)H * HS * sizeof(float);  // 69632 B
        eqdec_edge_mlp_scatter<<<blocks, 128, shmem, stream>>>(
            m_ij, x_, edge_idx, ew_W1, ew_b1, ew_W2, ew_b2, gsum, gcnt, E);
    }
    // 3) node MLP + velocity combine + geom delta (2 waves/block, 16 nodes/wave)
    {
        const int blocks = (N + 31) / 32;
        const size_t shmem = ((size_t)H * HS + 2 * 16 * HS) * sizeof(float);  // 87040 B
        eqdec_node_mlp_combine<<<blocks, 64, shmem, stream>>>(
            h_, vel_all, vg_W1, vg_b1, vg_W2, vg_b2, gsum, gcnt,
            (float*)d_out, N);
    }
}